// ReprojectLayer_batch_76871324664013
// MI455X (gfx1250) — compile-verified
//
#include <hip/hip_runtime.h>

// CDNA5 / gfx1250: wave32, WMMA 16x16x4 f32.
typedef __attribute__((ext_vector_type(2))) float v2f;
typedef __attribute__((ext_vector_type(8))) float v8f;

#define NPTS  1024
#define MCAM  4
#define NTILES (NPTS / 16)          // 64 tiles of 16 points
#define WAVES_PER_BLOCK 8
#define TILES_PER_WAVE (NTILES / WAVES_PER_BLOCK) // 8

// One block per sample b.
//  A (16x4 f32): lane half h=lane/16 selects K-pair {0,1} (x,y) vs {2,3} (z,1);
//                vgpr 0/1 select within the pair.  (ISA 7.12.2 table, 32-bit A 16x4)
//  B (4x16 f32): lane%16 = output column (camera m, proj row j -> col = m*3+j),
//                same K mapping as A (symmetric layout assumption).
//  D (16x16 f32): lane%16 = column, vgpr v + 8*h = point row within the tile.
__global__ __launch_bounds__(256) void reproject_wmma_kernel(
    const float* __restrict__ pts,   // (B, NPTS*3)
    const float* __restrict__ proj,  // (B, MCAM*12)
    float* __restrict__ out)         // (B, MCAM*NPTS*2)
{
    const int b    = blockIdx.x;
    const int tid  = threadIdx.x;
    const int wave = tid >> 5;
    const int lane = tid & 31;
    const int n    = lane & 15;   // column index / point-in-half
    const int h    = lane >> 4;   // half selects K pair

    const float* __restrict__ pbase = pts  + (size_t)b * (NPTS * 3);
    const float* __restrict__ qbase = proj + (size_t)b * (MCAM * 12);
    float*       __restrict__ obase = out  + (size_t)b * (MCAM * NPTS * 2);

    // ---- B matrix (projection rows), loaded once per wave ----
    const bool valid = (n < 12);
    const int  m = n / 3;             // camera
    const int  j = n % 3;             // proj row: 0->x, 1->y, 2->w
    v2f bm;
    if (valid) {
        const int off = m * 12 + j * 4 + 2 * h;   // proj[m][j][2h .. 2h+1]
        bm.x = qbase[off];
        bm.y = qbase[off + 1];
    } else {
        bm.x = 0.0f; bm.y = 0.0f;     // padding columns 12..15
    }

    // lane holding w for this lane's camera (same half-wave)
    const int waddr = ((m * 3 + 2) + (h << 4)) << 2;   // byte index for ds_bpermute

    // per-lane output offset (floats): out[m][p][c], p = tile*16 + v + 8h, c = j
    const bool do_store = valid && (j != 2);
    const int  laneoff  = m * (NPTS * 2) + h * 16 + j;

    #pragma unroll 2
    for (int t = 0; t < TILES_PER_WAVE; ++t) {
        const int tile = wave * TILES_PER_WAVE + t;
        const int p    = tile * 16 + n;
        const int idx  = p * 3 + 2 * h;

        // Branchless A-load: both loads unconditional and in-bounds.
        //   h=0: ld0 = x (p*3),   ld1 = y (p*3+1)
        //   h=1: ld0 = z (p*3+2), ld1 = y (p*3+1, discarded)
        const float ld0 = pbase[idx];
        const float ld1 = pbase[idx + 1 - 2 * h];
        v2f a;
        a.x = ld0;
        a.y = h ? 1.0f : ld1;                     // v_cndmask, no branch

        v8f acc = {0.f, 0.f, 0.f, 0.f, 0.f, 0.f, 0.f, 0.f};
        // D = A(16x4) * B(4x16) + 0  -> 12 homogeneous outputs x 16 points
        acc = __builtin_amdgcn_wmma_f32_16x16x4_f32(
            /*neg_a=*/false, a, /*neg_b=*/false, bm,
            /*c_mod=*/(short)0, acc, /*reuse_a=*/false, /*reuse_b=*/false);

        // Batch all 8 w-broadcasts (convergent, full-EXEC) -> one DScnt wait.
        float wv[8];
        #pragma unroll
        for (int v = 0; v < 8; ++v)
            wv[v] = __int_as_float(
                __builtin_amdgcn_ds_bpermute(waddr, __float_as_int(acc[v])));

        // Single predicated region per tile: divide + 8 stores.
        if (do_store) {
            float* op = obase + tile * 32 + laneoff;
            #pragma unroll
            for (int v = 0; v < 8; ++v) {
                float w = wv[v];
                float r = __builtin_amdgcn_rcpf(w);
                r = r * (2.0f - w * r);           // one Newton step
                op[v * 2] = acc[v] * r;           // 8B-stride; 128B/camera/tile
            }
        }
    }
}

extern "C" void kernel_launch(void* const* d_in, const int* in_sizes, int n_in,
                              void* d_out, int out_size, void* d_ws, size_t ws_size,
                              hipStream_t stream) {
    const float* pts  = (const float*)d_in[0];   // (B, N*3) f32
    const float* proj = (const float*)d_in[1];   // (B, M*3*4) f32
    float* out = (float*)d_out;                  // (B, M*N*2) f32

    const int B = in_sizes[0] / (NPTS * 3);
    reproject_wmma_kernel<<<dim3(B), dim3(32 * WAVES_PER_BLOCK), 0, stream>>>(pts, proj, out);
}